// E2E_DRO_Module_4312147165890
// MI455X (gfx1250) — compile-verified
//
#include <hip/hip_runtime.h>
#include <hip/hip_bf16.h>
#include <math.h>

// ---------------------------------------------------------------------------
// E2E DRO module for MI455X (gfx1250, wave32, WMMA, TDM).
//   N_OBS=1024, N_X=128, N_Y=256, 200 FISTA iterations.
// Pipeline:
//   k1: Y_hat = X W^T + b                 (WMMA f32 16x16x4, K=128, 2 chains)
//   k2: epc   = (Y - Y_hat) - colmean
//   k3: Sigma = epc^T epc / 1024          (WMMA f32 16x16x4, K=1024, 2 chains)
//   k4: step  = 1/(2*||Sigma||_F + 1e-8)
//   k5: persistent FISTA: 64 WGs x 16 rows; Sigma (256KB f32) staged into
//       320KB CDNA5 LDS via TENSOR_LOAD_TO_LDS (TDM) + s_wait_tensorcnt;
//       grad via dual-chain WMMA f32 from LDS (Sigma symmetric -> contiguous
//       b64 LDS reads, shared A fragment); bitonic sort + fused scan for the
//       simplex projection; 200 iters with zero global round trips.
// ---------------------------------------------------------------------------

typedef __attribute__((ext_vector_type(2))) float v2f;
typedef __attribute__((ext_vector_type(8))) float v8f;
typedef __attribute__((ext_vector_type(4))) unsigned int v4u;
typedef __attribute__((ext_vector_type(8))) int v8i;
typedef __attribute__((ext_vector_type(4))) int v4i;

#define N_OBS 1024
#define N_X   128
#define N_Y   256

__device__ __forceinline__ v8f wmma_f32x4(v2f a, v2f b, v8f c) {
  // (neg_a, A, neg_b, B, c_mod, C, reuse_a, reuse_b)
  return __builtin_amdgcn_wmma_f32_16x16x4_f32(false, a, false, b, (short)0, c,
                                               false, false);
}

// A 16x4 f32 fragment layout (ISA 7.12.2): lane L -> M = L%16,
// VGPR v holds K = 2*(L/16) + v (each lane loads a contiguous K-pair).
// B 4x16 mirrors with N = L%16. C/D 16x16: VGPR v -> M = v + 8*(L/16),
// N = L%16.

// ---------------- Kernel 1: Y_hat = X W^T + b ------------------------------
__global__ void __launch_bounds__(256)
yhat_kernel(const float* __restrict__ X, const float* __restrict__ W,
            const float* __restrict__ b, float* __restrict__ Yhat) {
  const int lane = threadIdx.x & 31;
  const int wave = blockIdx.x * 8 + (threadIdx.x >> 5);  // 1024 waves
  const int mt = wave >> 4;   // 0..63 (M tiles)
  const int nt = wave & 15;   // 0..15 (N tiles)
  const int e = lane & 15;
  const int half = lane >> 4;
  const int koff = half * 2;

  const int arow = mt * 16 + e;  // A row (X row)
  const int bcol = nt * 16 + e;  // B col (W row: B[k][n] = W[n*128+k])

  // Two independent accumulation chains to hide XDL latency.
  v8f acc0 = {0.f, 0.f, 0.f, 0.f, 0.f, 0.f, 0.f, 0.f};
  v8f acc1 = {0.f, 0.f, 0.f, 0.f, 0.f, 0.f, 0.f, 0.f};
  for (int kb = 0; kb < N_X; kb += 8) {
    v2f a0 = *(const v2f*)&X[arow * N_X + kb + koff];
    v2f b0 = *(const v2f*)&W[bcol * N_X + kb + koff];
    v2f a1 = *(const v2f*)&X[arow * N_X + kb + 4 + koff];
    v2f b1 = *(const v2f*)&W[bcol * N_X + kb + 4 + koff];
    acc0 = wmma_f32x4(a0, b0, acc0);
    acc1 = wmma_f32x4(a1, b1, acc1);
  }
  const float bias = b[nt * 16 + e];
#pragma unroll
  for (int vg = 0; vg < 8; ++vg) {
    const int m = mt * 16 + vg + 8 * half;
    Yhat[m * N_Y + nt * 16 + e] = acc0[vg] + acc1[vg] + bias;
  }
}

// ---------------- Kernel 2: epc = (Y - Yhat) - colmean ---------------------
__global__ void __launch_bounds__(256)
epc_kernel(const float* __restrict__ Y, const float* __restrict__ Yhat,
           float* __restrict__ epc) {
  const int j = threadIdx.x;  // one block, 256 threads, one column each
  float s = 0.f;
  for (int i = 0; i < N_OBS; ++i) s += Y[i * N_Y + j] - Yhat[i * N_Y + j];
  const float mu = s * (1.0f / (float)N_OBS);
  for (int i = 0; i < N_OBS; ++i)
    epc[i * N_Y + j] = Y[i * N_Y + j] - Yhat[i * N_Y + j] - mu;
}

// ---------------- Kernel 3: Sigma = epc^T epc / N_OBS ----------------------
__global__ void __launch_bounds__(256)
sigma_kernel(const float* __restrict__ epc, float* __restrict__ Sigma) {
  const int lane = threadIdx.x & 31;
  const int wave = blockIdx.x * 8 + (threadIdx.x >> 5);  // 256 waves
  const int pt = wave >> 4;
  const int qt = wave & 15;
  const int e = lane & 15;
  const int half = lane >> 4;
  const int koff = half * 2;
  const int p = pt * 16 + e;
  const int q = qt * 16 + e;

  v8f acc0 = {0.f, 0.f, 0.f, 0.f, 0.f, 0.f, 0.f, 0.f};
  v8f acc1 = {0.f, 0.f, 0.f, 0.f, 0.f, 0.f, 0.f, 0.f};
  for (int kb = 0; kb < N_OBS; kb += 8) {
    const int k0 = kb + koff;
    v2f a0 = {epc[k0 * N_Y + p], epc[(k0 + 1) * N_Y + p]};
    v2f b0 = {epc[k0 * N_Y + q], epc[(k0 + 1) * N_Y + q]};
    v2f a1 = {epc[(k0 + 4) * N_Y + p], epc[(k0 + 5) * N_Y + p]};
    v2f b1 = {epc[(k0 + 4) * N_Y + q], epc[(k0 + 5) * N_Y + q]};
    acc0 = wmma_f32x4(a0, b0, acc0);
    acc1 = wmma_f32x4(a1, b1, acc1);
  }
#pragma unroll
  for (int vg = 0; vg < 8; ++vg) {
    const int pm = pt * 16 + vg + 8 * half;
    Sigma[pm * N_Y + qt * 16 + e] = (acc0[vg] + acc1[vg]) * (1.0f / (float)N_OBS);
  }
}

// ---------------- Kernel 4: step = 1/(2*||Sigma||_F + 1e-8) ----------------
__global__ void __launch_bounds__(256)
step_kernel(const float* __restrict__ Sigma, float* __restrict__ step_out) {
  __shared__ float red[256];
  float s = 0.f;
  for (int i = threadIdx.x; i < N_Y * N_Y; i += 256) {
    const float v = Sigma[i];
    s += v * v;
  }
  red[threadIdx.x] = s;
  __syncthreads();
  for (int off = 128; off > 0; off >>= 1) {
    if (threadIdx.x < off) red[threadIdx.x] += red[threadIdx.x + off];
    __syncthreads();
  }
  if (threadIdx.x == 0) step_out[0] = 1.0f / (2.0f * sqrtf(red[0]) + 1e-8f);
}

// ---------------- Kernel 5: persistent FISTA -------------------------------
// Dynamic LDS layout (floats): sigma[65536] zy[4096] z[4096] v[4096] theta[32]
#define FISTA_LDS_FLOATS (65536 + 4096 * 3 + 32)

__global__ void __launch_bounds__(256)
fista_kernel(const float* __restrict__ Sigma, const float* __restrict__ Yhat,
             const float* __restrict__ step_p, float* __restrict__ Zout) {
  extern __shared__ float smem[];
  float* s_sigma = smem;            // 256 KB (LDS offset 0)
  float* s_zy    = smem + 65536;    // 16 KB (doubles as sort buffer)
  float* s_z     = s_zy + 4096;     // 16 KB
  float* s_v     = s_z + 4096;      // 16 KB
  float* s_theta = s_v + 4096;      // 16 thetas

  const int tid  = threadIdx.x;
  const int lane = tid & 31;
  const int wv   = tid >> 5;         // 0..7
  const int row0 = blockIdx.x * 16;  // 64 blocks x 16 rows
  const float step = step_p[0];

#if __has_builtin(__builtin_amdgcn_tensor_load_to_lds)
  // Stage Sigma (256KB) into LDS with one Tensor Data Mover op (wave 0 only).
  // D# group0: count=1 | lds_addr=0 | global_addr | type=2.
  // D# group1: data_size=4B | tensor_dim0/1=256 | tile_dim0/1=256 | stride0=256.
  if (wv == 0) {
    const unsigned long long ga = (unsigned long long)(uintptr_t)Sigma;
    v4u g0;
    g0[0] = 1u;                                   // count=1, user mode
    g0[1] = 0u;                                   // lds_addr = dyn-LDS base
    g0[2] = (unsigned int)ga;                     // global_addr[31:0]
    g0[3] = (unsigned int)((ga >> 32) & 0x01FFFFFFu) | 0x80000000u;  // type=2
    v8i g1;
    g1[0] = 0x20000;                              // data_size=2 (4 bytes)
    g1[1] = (int)(256u << 16);                    // tensor_dim0[15:0]
    g1[2] = (int)(256u << 16);                    // dim0[31:16]=0 | dim1[15:0]
    g1[3] = (int)(256u << 16);                    // dim1[31:16]=0 | tile_dim0
    g1[4] = 256;                                  // tile_dim1=256, tile_dim2=0
    g1[5] = 256;                                  // tensor_dim0_stride[31:0]
    g1[6] = 0;                                    // stride0[47:32] | stride1 lo
    g1[7] = 0;
    v4i gz4 = {0, 0, 0, 0};                       // groups 2/3 unused (2D)
    v8i gz8 = {0, 0, 0, 0, 0, 0, 0, 0};           // extra group (clang-23 arity)
    __builtin_amdgcn_tensor_load_to_lds(g0, g1, gz4, gz4, gz8, 0);
    __builtin_amdgcn_s_wait_tensorcnt(0);
  }
#else
  for (int i = tid * 4; i < N_Y * N_Y; i += 256 * 4)
    *(float4*)&s_sigma[i] = *(const float4*)&Sigma[i];
#endif
  for (int g = tid; g < 4096; g += 256) {
    s_z[g]  = 1.0f / (float)N_Y;
    s_zy[g] = 1.0f / (float)N_Y;
  }
  __syncthreads();

  const int e = lane & 15;
  const int half = lane >> 4;
  const int koff = half * 2;
  const int n0 = (wv * 2) * 16 + e;   // this wave's two N-tiles
  const int n1 = n0 + 16;

  // Preload this wave's Y_hat fragments (2 N-tiles, D layout) once.
  float yh0[8], yh1[8];
#pragma unroll
  for (int vg = 0; vg < 8; ++vg) {
    const int m = row0 + vg + 8 * half;
    yh0[vg] = Yhat[m * N_Y + n0];
    yh1[vg] = Yhat[m * N_Y + n1];
  }

  float t = 1.0f;
  for (int iter = 0; iter < 200; ++iter) {
    // ---- v = Zy - step * (2*Zy@Sigma - Yhat): dual-chain WMMA from LDS ---
    // A fragment (Zy row m=e) is shared by both N-tiles.
    v8f acc0 = {0.f, 0.f, 0.f, 0.f, 0.f, 0.f, 0.f, 0.f};
    v8f acc1 = {0.f, 0.f, 0.f, 0.f, 0.f, 0.f, 0.f, 0.f};
    for (int kb = 0; kb < N_Y; kb += 4) {
      v2f a  = *(const v2f*)&s_zy[e * N_Y + kb + koff];
      v2f b0 = *(const v2f*)&s_sigma[n0 * N_Y + kb + koff];  // Sigma symmetric
      v2f b1 = *(const v2f*)&s_sigma[n1 * N_Y + kb + koff];
      acc0 = wmma_f32x4(a, b0, acc0);
      acc1 = wmma_f32x4(a, b1, acc1);
    }
#pragma unroll
    for (int vg = 0; vg < 8; ++vg) {
      const int m = vg + 8 * half;
      s_v[m * N_Y + n0] = s_zy[m * N_Y + n0] - step * (2.0f * acc0[vg] - yh0[vg]);
      s_v[m * N_Y + n1] = s_zy[m * N_Y + n1] - step * (2.0f * acc1[vg] - yh1[vg]);
    }
    __syncthreads();

    // ---- simplex projection: sort copy of v descending per row -----------
    for (int g = tid; g < 4096; g += 256) s_zy[g] = s_v[g];
    __syncthreads();
    for (int k = 2; k <= 256; k <<= 1) {
      for (int j = k >> 1; j > 0; j >>= 1) {
        for (int g = tid; g < 4096; g += 256) {
          const int i = g & 255;
          const int ixj = i ^ j;
          if (ixj > i) {
            const int base = g & ~255;
            const float a0 = s_zy[base + i];
            const float a1 = s_zy[base + ixj];
            // overall-descending bitonic comparator
            const bool sw = ((i & k) == 0) ? (a0 < a1) : (a0 > a1);
            if (sw) { s_zy[base + i] = a1; s_zy[base + ixj] = a0; }
          }
        }
        __syncthreads();
      }
    }

    // ---- fused cumsum / support / theta scan (one thread per row) --------
    if (tid < 16) {
      float acc = 0.f, theta = 0.f;
      for (int jj = 0; jj < 256; ++jj) {
        const float u = s_zy[tid * N_Y + jj];
        acc += u;
        const float th = (acc - 1.0f) / (float)(jj + 1);
        if (u > th) theta = th;  // holds for a prefix; last true wins
      }
      s_theta[tid] = theta;
    }
    __syncthreads();

    // ---- Zn = max(v - theta, 0); momentum update -------------------------
    const float tn = 0.5f * (1.0f + sqrtf(1.0f + 4.0f * t * t));
    const float mom = (t - 1.0f) / tn;
    t = tn;
    for (int g = tid; g < 4096; g += 256) {
      const float zn0 = s_v[g] - s_theta[g >> 8];
      const float zn = zn0 > 0.0f ? zn0 : 0.0f;
      const float zold = s_z[g];
      s_z[g] = zn;
      s_zy[g] = zn + mom * (zn - zold);
    }
    __syncthreads();
  }

  for (int g = tid; g < 4096; g += 256)
    Zout[(row0 + (g >> 8)) * N_Y + (g & 255)] = s_z[g];
}

// ---------------------------------------------------------------------------
extern "C" void kernel_launch(void* const* d_in, const int* in_sizes, int n_in,
                              void* d_out, int out_size, void* d_ws,
                              size_t ws_size, hipStream_t stream) {
  const float* X = (const float*)d_in[0];  // 1024x128
  const float* Y = (const float*)d_in[1];  // 1024x256
  const float* W = (const float*)d_in[2];  // 256x128
  const float* b = (const float*)d_in[3];  // 256

  float* z_out = (float*)d_out;                    // 1024x256
  float* yhat  = (float*)d_out + N_OBS * N_Y;      // 1024x256 (2nd output)

  // Workspace: epc (1MB) | Sigma (256KB) | step (4B)  -> ~1.31 MB
  float* epc   = (float*)d_ws;
  float* Sigma = epc + N_OBS * N_Y;
  float* stepp = Sigma + N_Y * N_Y;

  // k1: 1024 tiles (64 M x 16 N), 8 waves/block
  yhat_kernel<<<128, 256, 0, stream>>>(X, W, b, yhat);
  // k2: column means + centered residuals
  epc_kernel<<<1, 256, 0, stream>>>(Y, yhat, epc);
  // k3: 256 tiles (16x16), 8 waves/block
  sigma_kernel<<<32, 256, 0, stream>>>(epc, Sigma);
  // k4: Frobenius norm -> step
  step_kernel<<<1, 256, 0, stream>>>(Sigma, stepp);
  // k5: persistent FISTA, 64 row-blocks, ~304 KB dynamic LDS per WG
  fista_kernel<<<64, 256, FISTA_LDS_FLOATS * sizeof(float), stream>>>(
      Sigma, yhat, stepp, z_out);
}